// BootGCNLayer_721554506533
// MI455X (gfx1250) — compile-verified
//
#include <hip/hip_runtime.h>
#include <hip/hip_bf16.h>

// ---------------------------------------------------------------------------
// BootGCN hyper-attention for MI455X (gfx1250, wave32).
// Dense projections run on V_WMMA_F32_16X16X4_F32 (fp32-exact, matches the
// fp32 reference through exp/softmax/LayerNorm). One wave computes a 16x64
// output strip (4 WMMA accumulators) so the A fragment is loaded once per
// k-step: steady state 4 wmma : 5 b64 loads. Edge softmax + scatter use
// VALU + global float atomics (edge tables stay L2-resident: 192MB L2).
// ---------------------------------------------------------------------------

typedef float v2f __attribute__((ext_vector_type(2)));
typedef float v8f __attribute__((ext_vector_type(8)));

#define FULL_BLOCKS 2048
#define TPB 256

// Monotonic float<->int key for atomicMax on signed ints.
__device__ __forceinline__ int fenc(float f) {
    int i = __float_as_int(f);
    return (i >= 0) ? i : (i ^ 0x7FFFFFFF);
}
__device__ __forceinline__ float fdec(int k) {
    return __int_as_float((k >= 0) ? k : (k ^ 0x7FFFFFFF));
}

// ---------------------------------------------------------------------------
// C[M x N] = A[M x K] @ W^T + bias      (W is [N x K] row-major, all fp32)
// One wave computes a 16x64 strip (four 16x16 WMMA tiles sharing the A rows).
// A 16x4 layout: lanes 0-15 = rows M; vgpr0={K0|K2}, vgpr1={K1|K3} (half split)
// B 4x16 layout: lanes 0-15 = cols N; same K split  (B[k][n] = W[n][k])
// D 16x16: vgpr i -> row tileM+i (lanes 0-15) / tileM+i+8 (lanes 16-31)
// Requires N % 64 == 0 (we pad DK=50 -> 64).
// ---------------------------------------------------------------------------
__global__ void gemm_wmma_f32(const float* __restrict__ A,
                              const float* __restrict__ W,
                              const float* __restrict__ bias,
                              float* __restrict__ C,
                              int M, int N, int K) {
    const int lane = threadIdx.x & 31;
    const int wave = (int)(((size_t)blockIdx.x * blockDim.x + threadIdx.x) >> 5);
    const int groupsN = N >> 6;                   // 64-wide strips
    const int tilesM  = (M + 15) >> 4;
    if (wave >= tilesM * groupsN) return;         // wave-uniform: EXEC stays all-1
    const int tileM  = (wave / groupsN) << 4;
    const int tileN0 = (wave % groupsN) << 6;
    const int half = lane >> 4;                   // 0: K{0,1}, 1: K{2,3}
    const int l16  = lane & 15;

    int ar = tileM + l16; if (ar >= M) ar = M - 1;   // clamp loads only
    const float* arow = A + (size_t)ar * K;
    const float* wrow[4];
#pragma unroll
    for (int j = 0; j < 4; ++j)
        wrow[j] = W + (size_t)(tileN0 + 16 * j + l16) * K;

    v8f acc[4] = {};
    for (int k0 = 0; k0 < K; k0 += 4) {
        const int ka = k0 + (half << 1);
        const float2 av = *(const float2*)(arow + ka);
        v2f a; a.x = av.x; a.y = av.y;
#pragma unroll
        for (int j = 0; j < 4; ++j) {
            const float2 bv = *(const float2*)(wrow[j] + ka);
            v2f b; b.x = bv.x; b.y = bv.y;
            acc[j] = __builtin_amdgcn_wmma_f32_16x16x4_f32(
                false, a, false, b, (short)0, acc[j], false, false);
        }
    }
#pragma unroll
    for (int j = 0; j < 4; ++j) {
        const int colj = tileN0 + 16 * j + l16;
        const float bn = bias[colj];
#pragma unroll
        for (int i = 0; i < 8; ++i) {
            const int row = tileM + i + (half << 3);
            if (row < M)
                C[(size_t)row * N + colj] = acc[j][i] + bn;
        }
    }
}

// Pad W[50x128] -> Wp[64x128] (zero rows 50..63), bias likewise.
__global__ void pad_w(const float* __restrict__ W, const float* __restrict__ b,
                      float* __restrict__ Wp, float* __restrict__ bp,
                      int N, int Np, int K) {
    int tid = blockIdx.x * blockDim.x + threadIdx.x;
    if (tid < Np * K) {
        int row = tid / K;
        Wp[tid] = (row < N) ? W[tid] : 0.0f;
    }
    if (tid < Np) bp[tid] = (tid < N) ? b[tid] : 0.0f;
}

// gq[n] = seeds · w_gs[n] + b_gs[n], n<50 (padded to 64 with zeros). 1 block.
__global__ void compute_gq(const float* __restrict__ seeds,
                           const float* __restrict__ wgs,
                           const float* __restrict__ bgs,
                           float* __restrict__ gq, int Kg) {
    const int lane = threadIdx.x & 31;
    const int wid  = threadIdx.x >> 5;
    for (int n = wid; n < 64; n += 8) {
        float v = 0.0f;
        if (n < 50) {
            const float* wr = wgs + (size_t)n * Kg;
            float p = 0.0f;
            for (int k = lane; k < Kg; k += 32) p += seeds[k] * wr[k];
#pragma unroll
            for (int o = 16; o; o >>= 1) p += __shfl_xor(p, o, 32);
            v = p + bgs[n];
        }
        if (lane == 0) gq[n] = v;
    }
}

// gk[j] = k[j] (64, zero-padded) · gq (64, zero-padded). One wave per node.
__global__ void compute_gk(const float* __restrict__ kk,
                           const float* __restrict__ gq,
                           float* __restrict__ gk, int Mk) {
    const int lane = threadIdx.x & 31;
    const int gw = (int)(((size_t)blockIdx.x * blockDim.x + threadIdx.x) >> 5);
    const int nw = (gridDim.x * blockDim.x) >> 5;
    const float2 q = ((const float2*)gq)[lane];
    for (int j = gw; j < Mk; j += nw) {
        const float2 a = ((const float2*)(kk + (size_t)j * 64))[lane];
        float p = a.x * q.x + a.y * q.y;
#pragma unroll
        for (int o = 16; o; o >>= 1) p += __shfl_xor(p, o, 32);
        if (lane == 0) gk[j] = p;
    }
}

__global__ void init_gmax(int* g) {
    if (threadIdx.x == 0) { g[0] = (int)0x80000000; g[1] = (int)0x80000000; }
}

// Pass 1: per-edge local score (64-wide dot) + global score; track global max
// hierarchically (wave -> block -> 2 atomics/block).
__global__ void edge_scores(const int* __restrict__ rows,
                            const int* __restrict__ cols, int E,
                            const float* __restrict__ lq,
                            const float* __restrict__ kk,
                            const float* __restrict__ gk, float inv_temp,
                            float* __restrict__ lsc, float* __restrict__ gsc,
                            int* __restrict__ gmax) {
    __shared__ int smaxl[8], smaxg[8];
    const int lane = threadIdx.x & 31;
    const int wid  = threadIdx.x >> 5;
    const int gw = (int)(((size_t)blockIdx.x * blockDim.x + threadIdx.x) >> 5);
    const int nw = (gridDim.x * blockDim.x) >> 5;
    float lml = -3.4e38f, lmg = -3.4e38f;
    for (int e = gw; e < E; e += nw) {
        const int r = rows[e], c = cols[e];
        const float2 a = ((const float2*)(lq + (size_t)r * 64))[lane];
        const float2 b = ((const float2*)(kk + (size_t)c * 64))[lane];
        float p = a.x * b.x + a.y * b.y;
#pragma unroll
        for (int o = 16; o; o >>= 1) p += __shfl_xor(p, o, 32);
        const float ls = p * inv_temp;
        const float gs = gk[c] * inv_temp;
        if (lane == 0) { lsc[e] = ls; gsc[e] = gs; }
        lml = fmaxf(lml, ls); lmg = fmaxf(lmg, gs);
    }
    if (lane == 0) { smaxl[wid] = fenc(lml); smaxg[wid] = fenc(lmg); }
    __syncthreads();
    if (threadIdx.x == 0) {
        int ml = smaxl[0], mg = smaxg[0];
        for (int i = 1; i < 8; ++i) { ml = max(ml, smaxl[i]); mg = max(mg, smaxg[i]); }
        atomicMax(&gmax[0], ml);
        atomicMax(&gmax[1], mg);
    }
}

// Pass 2: exponentiate (stability shift by global max) + per-row denominators.
__global__ void edge_exp(const int* __restrict__ rows, int E,
                         const int* __restrict__ gmax,
                         float* __restrict__ lsc, float* __restrict__ gsc,
                         float* __restrict__ den_l, float* __restrict__ den_g) {
    const float ml = fdec(gmax[0]);
    const float mg = fdec(gmax[1]);
    for (int e = blockIdx.x * blockDim.x + threadIdx.x; e < E;
         e += gridDim.x * blockDim.x) {
        const float el = __expf(lsc[e] - ml);
        const float eg = __expf(gsc[e] - mg);
        lsc[e] = el; gsc[e] = eg;
        const int r = rows[e];
        atomicAdd(&den_l[r], el);
        atomicAdd(&den_g[r], eg);
    }
}

// Pass 3: out[r] += attn_e * v[c]; one wave per edge, 4 features per lane.
__global__ void edge_aggregate(const int* __restrict__ rows,
                               const int* __restrict__ cols, int E,
                               const float* __restrict__ lsc,
                               const float* __restrict__ gsc,
                               const float* __restrict__ den_l,
                               const float* __restrict__ den_g,
                               const float* __restrict__ vv,
                               float* __restrict__ out) {
    const int lane = threadIdx.x & 31;
    const int gw = (int)(((size_t)blockIdx.x * blockDim.x + threadIdx.x) >> 5);
    const int nw = (gridDim.x * blockDim.x) >> 5;
    for (int e = gw; e < E; e += nw) {
        const int r = rows[e], c = cols[e];
        const float attn = 0.5f * (lsc[e] / den_l[r] + gsc[e] / den_g[r]);
        const float4 v4 = ((const float4*)(vv + (size_t)c * 128))[lane];
        float* o = out + (size_t)r * 128 + lane * 4;
        atomicAdd(o + 0, attn * v4.x);
        atomicAdd(o + 1, attn * v4.y);
        atomicAdd(o + 2, attn * v4.z);
        atomicAdd(o + 3, attn * v4.w);
    }
}

// Pass 4: out = LayerNorm(relu(out + residual)) * ln_w + ln_b, wave per row.
__global__ void relu_ln(float* __restrict__ out, const float* __restrict__ res,
                        const float* __restrict__ lnw,
                        const float* __restrict__ lnb, int M) {
    const int lane = threadIdx.x & 31;
    const int gw = (int)(((size_t)blockIdx.x * blockDim.x + threadIdx.x) >> 5);
    const int nw = (gridDim.x * blockDim.x) >> 5;
    for (int r = gw; r < M; r += nw) {
        float4 o  = ((const float4*)(out + (size_t)r * 128))[lane];
        const float4 rr = ((const float4*)(res + (size_t)r * 128))[lane];
        o.x = fmaxf(o.x + rr.x, 0.0f);
        o.y = fmaxf(o.y + rr.y, 0.0f);
        o.z = fmaxf(o.z + rr.z, 0.0f);
        o.w = fmaxf(o.w + rr.w, 0.0f);
        float s  = o.x + o.y + o.z + o.w;
        float s2 = o.x * o.x + o.y * o.y + o.z * o.z + o.w * o.w;
#pragma unroll
        for (int off = 16; off; off >>= 1) {
            s  += __shfl_xor(s, off, 32);
            s2 += __shfl_xor(s2, off, 32);
        }
        const float mu  = s * (1.0f / 128.0f);
        const float var = s2 * (1.0f / 128.0f) - mu * mu;
        const float rstd = rsqrtf(var + 1e-5f);
        const float4 w4 = ((const float4*)lnw)[lane];
        const float4 b4 = ((const float4*)lnb)[lane];
        o.x = (o.x - mu) * rstd * w4.x + b4.x;
        o.y = (o.y - mu) * rstd * w4.y + b4.y;
        o.z = (o.z - mu) * rstd * w4.z + b4.z;
        o.w = (o.w - mu) * rstd * w4.w + b4.w;
        ((float4*)(out + (size_t)r * 128))[lane] = o;
    }
}

static inline void launch_gemm(const float* A, const float* W, const float* b,
                               float* C, int M, int N, int K, hipStream_t s) {
    long waves = (long)((M + 15) / 16) * (N / 64);
    int blocks = (int)((waves + 7) / 8);
    gemm_wmma_f32<<<blocks, TPB, 0, s>>>(A, W, b, C, M, N, K);
}

extern "C" void kernel_launch(void* const* d_in, const int* in_sizes, int n_in,
                              void* d_out, int out_size, void* d_ws, size_t ws_size,
                              hipStream_t stream) {
    const float* seeds   = (const float*)d_in[0];
    const float* e_input = (const float*)d_in[1];
    const float* p_input = (const float*)d_in[2];
    const int*   pe_adj  = (const int*)d_in[3];
    const int*   ep_adj  = (const int*)d_in[4];
    const int NE   = in_sizes[1] / 128;
    const int NP   = in_sizes[2] / 128;
    const int E_pe = in_sizes[3] / 2;
    const int E_ep = in_sizes[4] / 2;
    const int Kg   = in_sizes[0];            // N_SEED*D = 2048

    float* e_out = (float*)d_out;
    float* p_out = e_out + (size_t)NE * 128;

    // ---- workspace carve (fp32) ----
    float* f = (float*)d_ws;
    const size_t maxN = (size_t)(NP > NE ? NP : NE);
    const size_t maxE = (size_t)(E_pe > E_ep ? E_pe : E_ep);
    float* lq    = f; f += maxN * 64;
    float* kk    = f; f += maxN * 64;
    float* vv    = f; f += maxN * 128;
    float* res   = f; f += maxN * 128;
    float* lsc   = f; f += maxE;
    float* gsc   = f; f += maxE;
    float* den_l = f; f += maxN;
    float* den_g = f; f += maxN;
    float* gk    = f; f += maxN;
    float* gq    = f; f += 64;
    float* Wls   = f; f += 64 * 128;
    float* Wks   = f; f += 64 * 128;
    float* bls   = f; f += 64;
    float* bks   = f; f += 64;
    int*   gmax  = (int*)f; f += 4;
    if ((size_t)((char*)f - (char*)d_ws) > ws_size) return;

    const float inv_temp = 0.1414213562f;    // 1/sqrt(DK=50)

    auto stage = [&](const float* localq, int Mq, const float* kin, int Mk,
                     const int* rows, const int* cols, int E, int pb,
                     float* out) {
        const float* w_ls  = (const float*)d_in[pb + 0];
        const float* b_ls  = (const float*)d_in[pb + 1];
        const float* w_ks  = (const float*)d_in[pb + 2];
        const float* b_ks  = (const float*)d_in[pb + 3];
        const float* w_vs  = (const float*)d_in[pb + 4];
        const float* b_vs  = (const float*)d_in[pb + 5];
        const float* w_res = (const float*)d_in[pb + 6];
        const float* b_res = (const float*)d_in[pb + 7];
        const float* w_gs  = (const float*)d_in[pb + 8];
        const float* b_gs  = (const float*)d_in[pb + 9];
        const float* ln_w  = (const float*)d_in[pb + 10];
        const float* ln_b  = (const float*)d_in[pb + 11];

        pad_w<<<(64 * 128 + TPB - 1) / TPB, TPB, 0, stream>>>(w_ls, b_ls, Wls, bls, 50, 64, 128);
        pad_w<<<(64 * 128 + TPB - 1) / TPB, TPB, 0, stream>>>(w_ks, b_ks, Wks, bks, 50, 64, 128);

        launch_gemm(localq, Wls,   bls,   lq,  Mq, 64, 128, stream);   // lq
        launch_gemm(kin,    Wks,   bks,   kk,  Mk, 64, 128, stream);   // k
        launch_gemm(kin,    w_vs,  b_vs,  vv,  Mk, 128, 128, stream);  // v
        launch_gemm(localq, w_res, b_res, res, Mq, 128, 128, stream);  // residual

        compute_gq<<<1, TPB, 0, stream>>>(seeds, w_gs, b_gs, gq, Kg);
        compute_gk<<<512, TPB, 0, stream>>>(kk, gq, gk, Mk);

        init_gmax<<<1, 32, 0, stream>>>(gmax);
        hipMemsetAsync(den_l, 0, (size_t)Mq * sizeof(float), stream);
        hipMemsetAsync(den_g, 0, (size_t)Mq * sizeof(float), stream);

        edge_scores<<<FULL_BLOCKS, TPB, 0, stream>>>(rows, cols, E, lq, kk, gk,
                                                     inv_temp, lsc, gsc, gmax);
        edge_exp<<<FULL_BLOCKS, TPB, 0, stream>>>(rows, E, gmax, lsc, gsc,
                                                  den_l, den_g);
        hipMemsetAsync(out, 0, (size_t)Mq * 128 * sizeof(float), stream);
        edge_aggregate<<<FULL_BLOCKS, TPB, 0, stream>>>(rows, cols, E, lsc, gsc,
                                                        den_l, den_g, vv, out);
        relu_ln<<<FULL_BLOCKS, TPB, 0, stream>>>(out, res, ln_w, ln_b, Mq);
    };

    // Stage A: e2p — queries = phrases (rows of pe_adj), keys/values = entities.
    stage(p_input, NP, e_input, NE, pe_adj, pe_adj + E_pe, E_pe, 5, p_out);
    // Stage B: p2e — queries = entities, keys/values = p_out (stage A result).
    stage(e_input, NE, p_out, NP, ep_adj, ep_adj + E_ep, E_ep, 17, e_out);
}